// RGBToHVI_2044404433651
// MI455X (gfx1250) — compile-verified
//
#include <hip/hip_runtime.h>

// RGB -> HVI elementwise transform, layout (N, 3, H, W), H*W = 2^20.
// Memory-bound: 402 MB streamed once => non-temporal b128 loads/stores.
// ALU minimized via raw gfx1250 transcendentals:
//   v_sin_f32/v_cos_f32 take revolutions (sin(2*pi*x)), so both trig sites
//   need zero range reduction; pow(s,k) = v_exp_f32(k * v_log_f32(s)).

typedef float v4f __attribute__((ext_vector_type(4)));

#define HVI_EPS 1e-8f

__device__ __forceinline__ void hvi_pixel(float r, float g, float b, float k,
                                          float& X, float& Y, float& Z) {
    float v  = fmaxf(r, fmaxf(g, b));
    float mn = fminf(r, fminf(g, b));
    // one fast reciprocal reused for all three hue candidates
    float inv = __builtin_amdgcn_rcpf(v - mn + HVI_EPS);

    // jnp.where chain, lowest priority first:
    //   hue = where(b==v, hue_b, 0); where(g==v, hue_g, .);
    //         where(r==v, hue_r, .); where(mn==v, 0, .)
    float hue = (b == v) ? 4.0f + (r - g) * inv : 0.0f;
    hue       = (g == v) ? 2.0f + (b - r) * inv : hue;
    float xr  = (g - b) * inv;
    xr        = xr - 6.0f * floorf(xr * (1.0f / 6.0f));   // Python-style x % 6.0
    hue       = (r == v) ? xr : hue;
    hue       = (mn == v) ? 0.0f : hue;
    hue      *= (1.0f / 6.0f);          // hue in [0,1) revolutions

    float sat = (v == 0.0f) ? 0.0f
                            : (v - mn) * __builtin_amdgcn_rcpf(v + HVI_EPS);

    // color_sensitive = (sin(v*pi/2) + eps)^k
    // sin(v*pi/2) = sin(2*pi * v/4) -> v_sin_f32(v * 0.25)
    float s   = __builtin_amdgcn_sinf(v * 0.25f) + HVI_EPS;   // s > 0 always
    float cs  = __builtin_amdgcn_exp2f(k * __builtin_amdgcn_logf(s));

    // cos/sin(2*pi*hue) -> v_cos/v_sin on the revolution fraction directly
    float cx  = __builtin_amdgcn_cosf(hue);
    float cy  = __builtin_amdgcn_sinf(hue);

    float csat = cs * sat;
    X = csat * cx;
    Y = csat * cy;
    Z = v;
}

__global__ __launch_bounds__(256) void rgb2hvi_kernel(
    const float* __restrict__ img,   // (N, 3, HW) planar
    const float* __restrict__ kptr,  // density_k scalar
    float* __restrict__ out,         // (N, 3, HW) planar: X, Y, Z
    long long nvec)                  // total pixels / 4
{
    long long t = (long long)blockIdx.x * (long long)blockDim.x + threadIdx.x;
    if (t >= nvec) return;

    const long long HW = 1ll << 20;          // 1024*1024, divisible by 4
    long long pix  = t << 2;                 // first pixel of this float4
    long long bimg = pix >> 20;              // batch index
    long long p    = pix & (HW - 1);         // offset inside plane

    const float* ibase = img + bimg * 3 * HW + p;
    float*       obase = out + bimg * 3 * HW + p;

    // Streaming loads: th=NT so the 402MB working set doesn't thrash L2.
    v4f r4 = __builtin_nontemporal_load((const v4f*)(ibase));
    v4f g4 = __builtin_nontemporal_load((const v4f*)(ibase + HW));
    v4f b4 = __builtin_nontemporal_load((const v4f*)(ibase + 2 * HW));

    float k = kptr[0];                       // uniform -> scalar load, cached

    v4f X, Y, Z;
#pragma unroll
    for (int i = 0; i < 4; ++i) {
        float x, y, z;
        hvi_pixel(r4[i], g4[i], b4[i], k, x, y, z);
        X[i] = x; Y[i] = y; Z[i] = z;
    }

    __builtin_nontemporal_store(X, (v4f*)(obase));
    __builtin_nontemporal_store(Y, (v4f*)(obase + HW));
    __builtin_nontemporal_store(Z, (v4f*)(obase + 2 * HW));
}

extern "C" void kernel_launch(void* const* d_in, const int* in_sizes, int n_in,
                              void* d_out, int out_size, void* d_ws, size_t ws_size,
                              hipStream_t stream) {
    const float* img  = (const float*)d_in[0];   // (16,3,1024,1024) fp32
    const float* kptr = (const float*)d_in[1];   // (1,) fp32
    float*       out  = (float*)d_out;           // (16,3,1024,1024) fp32

    long long total = (long long)in_sizes[0];    // N * 3 * HW
    long long npix  = total / 3;                 // N * HW
    long long nvec  = npix >> 2;                 // float4 granules

    const int block = 256;                       // 8 wave32 waves per block
    long long grid  = (nvec + block - 1) / block;

    rgb2hvi_kernel<<<(unsigned int)grid, block, 0, stream>>>(img, kptr, out, nvec);
}